// L2Prompt_18519898981055
// MI455X (gfx1250) — compile-verified
//
#include <hip/hip_runtime.h>
#include <math.h>

#define LDIM 1250
#define PDIM 2048
#define BDIM 4096
#define TOPK 5
#define EPSV 1e-8f

typedef __attribute__((ext_vector_type(2))) float v2f;
typedef __attribute__((ext_vector_type(8))) float v8f;

// ---------------- workspace layout (floats) ----------------
// ws[0 .. PDIM-1]  : key row norms
// ws[PDIM]         : sum of top-5 scores (atomic)
// ws[PDIM+1]       : sum of per-row entropies (atomic)

__global__ void key_norm_kernel(const float* __restrict__ keys,
                                float* __restrict__ ws) {
  __shared__ float red[32];
  const int p = blockIdx.x;
  const int lane = threadIdx.x;
  const float* row = keys + (size_t)p * LDIM;
  float s = 0.f;
  for (int l = lane; l < LDIM; l += 32) { float v = row[l]; s += v * v; }
  red[lane] = s;
  __syncthreads();
  if (lane == 0) {
    float t = 0.f;
#pragma unroll
    for (int i = 0; i < 32; ++i) t += red[i];
    ws[p] = sqrtf(t);
    if (p == 0) { ws[PDIM] = 0.f; ws[PDIM + 1] = 0.f; }
  }
}

// LDS carve sizes (in floats / ints)
#define NS_SCORE (16 * PDIM)   // 32768 f
#define NS_NK    (PDIM)        // 2048 f
#define SMEM_FLOATS (NS_SCORE + NS_NK + 16 + 16 + 512 + 512 + 512 * TOPK)
#define SMEM_INTS   (512 * TOPK + 16 * TOPK)
#define SMEM_BYTES  ((size_t)(SMEM_FLOATS + SMEM_INTS) * 4)

__global__ __launch_bounds__(512) void fused_prompt_kernel(
    const float* __restrict__ ppg, const float* __restrict__ keys,
    const float* __restrict__ prompt, float* __restrict__ ws,
    float* __restrict__ out) {
  extern __shared__ char smem[];
  float* s_score  = (float*)smem;                 // [16][PDIM]
  float* s_nk     = s_score + NS_SCORE;           // [PDIM]
  float* s_nx     = s_nk + NS_NK;                 // [16]
  float* s_rowmax = s_nx + 16;                    // [16]
  float* s_partA  = s_rowmax + 16;                // [512]
  float* s_partB  = s_partA + 512;                // [512]
  float* s_topv   = s_partB + 512;                // [512*5]
  int*   s_topi   = (int*)(s_topv + 512 * TOPK);  // [512*5]
  int*   s_idx    = s_topi + 512 * TOPK;          // [16*5]

  const int tid  = threadIdx.x;
  const int lane = tid & 31;
  const int wave = tid >> 5;        // 0..15
  const int lr   = lane & 15;
  const int h    = lane >> 4;       // half-select (K pair)
  const int rowBase = blockIdx.x * 16;

  // ---- stage key norms into LDS ----
  for (int j = tid; j < PDIM; j += 512) s_nk[j] = ws[j];

  // ---- x row norms: wave w sums row w ----
  {
    const float* xr = ppg + (size_t)(rowBase + wave) * LDIM;
    float s = 0.f;
    for (int l = lane; l < LDIM; l += 32) { float v = xr[l]; s += v * v; }
    s_partA[tid] = s;
  }
  __syncthreads();
  if (tid < 16) {
    float t = 0.f;
    for (int i = 0; i < 32; ++i) t += s_partA[tid * 32 + i];
    s_nx[tid] = sqrtf(t);
  }
  __syncthreads();

  // ---- f32 WMMA GEMM: 16x16 tile per wave x 8 column tiles ----
  const float* xrow = ppg + (size_t)(rowBase + lr) * LDIM;
  const float* krow[8];
#pragma unroll
  for (int t = 0; t < 8; ++t)
    krow[t] = keys + (size_t)((wave * 8 + t) * 16 + lr) * LDIM;

  v8f acc[8];
  const v8f vzero = {0.f, 0.f, 0.f, 0.f, 0.f, 0.f, 0.f, 0.f};
#pragma unroll
  for (int t = 0; t < 8; ++t) acc[t] = vzero;

  const int koff = 2 * h;
  for (int k0 = 0; k0 + 4 <= LDIM; k0 += 4) {
    __builtin_prefetch(xrow + k0 + 64, 0, 1);
    v2f a = *(const v2f*)(xrow + k0 + koff);
#pragma unroll
    for (int t = 0; t < 8; ++t) {
      v2f b = *(const v2f*)(krow[t] + k0 + koff);
      acc[t] = __builtin_amdgcn_wmma_f32_16x16x4_f32(
          false, a, false, b, (short)0, acc[t], false, false);
    }
  }
  // tail: k = 1248,1249 (lanes 16..31 contribute zeros)
  {
    v2f a;
    if (h == 0) { a.x = xrow[1248]; a.y = xrow[1249]; }
    else        { a.x = 0.f;        a.y = 0.f; }
#pragma unroll
    for (int t = 0; t < 8; ++t) {
      v2f b;
      if (h == 0) { b.x = krow[t][1248]; b.y = krow[t][1249]; }
      else        { b.x = 0.f;           b.y = 0.f; }
      acc[t] = __builtin_amdgcn_wmma_f32_16x16x4_f32(
          false, a, false, b, (short)0, acc[t], false, false);
    }
  }

  // ---- fused cosine-distance, write 16x2048 stripe to LDS ----
  // C/D layout: VGPR r -> (M=r, N=lane) lanes 0-15 ; (M=r+8, N=lane-16)
#pragma unroll
  for (int t = 0; t < 8; ++t) {
    const int n = (wave * 8 + t) * 16 + lr;
    const float nk = s_nk[n];
#pragma unroll
    for (int r = 0; r < 8; ++r) {
      const int m = r + 8 * h;
      float num = acc[t][r];
      float den = fmaxf(s_nx[m] * nk, EPSV);
      s_score[m * PDIM + n] = 1.0f - num / den;
    }
  }
  __syncthreads();

  // ---- phase 2: per-row max / softmax stats / top-5 ----
  const int row = tid >> 5;   // 0..15
  const int sub = tid & 31;
  const float* srow = s_score + row * PDIM;

  {
    float m = -3.4e38f;
    for (int j = sub; j < PDIM; j += 32) m = fmaxf(m, srow[j]);
    s_partA[tid] = m;
  }
  __syncthreads();
  if (tid < 16) {
    float m = -3.4e38f;
    for (int i = 0; i < 32; ++i) m = fmaxf(m, s_partA[tid * 32 + i]);
    s_rowmax[tid] = m;
  }
  __syncthreads();

  {
    const float M = s_rowmax[row];
    float E = 0.f, SE = 0.f;
    float tv[TOPK]; int ti[TOPK];
#pragma unroll
    for (int i = 0; i < TOPK; ++i) { tv[i] = 3.4e38f; ti[i] = 0; }
    for (int j = sub; j < PDIM; j += 32) {
      float s = srow[j];
      float e = expf(s - M);
      E += e; SE += s * e;
      if (s < tv[TOPK - 1]) {
        int pos = TOPK - 1;
        while (pos > 0 && s < tv[pos - 1]) {
          tv[pos] = tv[pos - 1]; ti[pos] = ti[pos - 1]; --pos;
        }
        tv[pos] = s; ti[pos] = j;
      }
    }
    s_partA[tid] = E; s_partB[tid] = SE;
#pragma unroll
    for (int i = 0; i < TOPK; ++i) {
      s_topv[tid * TOPK + i] = tv[i];
      s_topi[tid * TOPK + i] = ti[i];
    }
  }
  __syncthreads();

  if (tid < 16) {
    float E = 0.f, SE = 0.f;
    for (int i = 0; i < 32; ++i) {
      E += s_partA[tid * 32 + i];
      SE += s_partB[tid * 32 + i];
    }
    float tv[TOPK]; int ti[TOPK];
#pragma unroll
    for (int i = 0; i < TOPK; ++i) { tv[i] = 3.4e38f; ti[i] = 0; }
    const int base = tid * 32 * TOPK;
    for (int c = 0; c < 32 * TOPK; ++c) {
      float v = s_topv[base + c];
      if (v < tv[TOPK - 1]) {
        int id = s_topi[base + c];
        int pos = TOPK - 1;
        while (pos > 0 && v < tv[pos - 1]) {
          tv[pos] = tv[pos - 1]; ti[pos] = ti[pos - 1]; --pos;
        }
        tv[pos] = v; ti[pos] = id;
      }
    }
    float logZ = s_rowmax[tid] + logf(E);
    float ent  = logZ - SE / E;
    float tsum = tv[0] + tv[1] + tv[2] + tv[3] + tv[4];
    atomicAdd(&ws[PDIM], tsum);
    atomicAdd(&ws[PDIM + 1], ent);
#pragma unroll
    for (int i = 0; i < TOPK; ++i) s_idx[tid * TOPK + i] = ti[i];
  }
  __syncthreads();

  // ---- phase 3: gather prompt rows, mean, add to ppg ----
  for (int m = 0; m < 16; ++m) {
    const int i0 = s_idx[m * TOPK + 0];
    const int i1 = s_idx[m * TOPK + 1];
    const int i2 = s_idx[m * TOPK + 2];
    const int i3 = s_idx[m * TOPK + 3];
    const int i4 = s_idx[m * TOPK + 4];
    const float* pb = ppg + (size_t)(rowBase + m) * LDIM;
    float* ob = out + (size_t)(rowBase + m) * LDIM;
    for (int c = tid; c < LDIM; c += 512) {
      float s = prompt[(size_t)i0 * LDIM + c] + prompt[(size_t)i1 * LDIM + c] +
                prompt[(size_t)i2 * LDIM + c] + prompt[(size_t)i3 * LDIM + c] +
                prompt[(size_t)i4 * LDIM + c];
      ob[c] = pb[c] + s * 0.2f;   // GLOBAL_COEFF=1.0, mean over 5
    }
  }
}

__global__ void finalize_kernel(const float* __restrict__ ws,
                                float* __restrict__ out) {
  if (threadIdx.x == 0 && blockIdx.x == 0) {
    out[(size_t)BDIM * LDIM]     = ws[PDIM] / (float)(BDIM * TOPK);
    out[(size_t)BDIM * LDIM + 1] = ws[PDIM + 1] / (float)BDIM;
  }
}

extern "C" void kernel_launch(void* const* d_in, const int* in_sizes, int n_in,
                              void* d_out, int out_size, void* d_ws,
                              size_t ws_size, hipStream_t stream) {
  const float* ppg    = (const float*)d_in[0];
  const float* keys   = (const float*)d_in[1];
  const float* prompt = (const float*)d_in[2];
  // d_in[3] = k (always 5, hardcoded as TOPK)
  float* out = (float*)d_out;
  float* ws  = (float*)d_ws;

  key_norm_kernel<<<PDIM, 32, 0, stream>>>(keys, ws);
  fused_prompt_kernel<<<BDIM / 16, 512, SMEM_BYTES, stream>>>(
      ppg, keys, prompt, ws, out);
  finalize_kernel<<<1, 1, 0, stream>>>(ws, out);
}